// Global_DuelForm_WSVectorQuantizer_79894981640747
// MI455X (gfx1250) — compile-verified
//
#include <hip/hip_runtime.h>
#include <hip/hip_bf16.h>
#include <math.h>

// ---------------------------------------------------------------------------
// DuelForm WS Vector Quantizer, gfx1250 (MI455X, wave32, WMMA + TDM).
//
// cost matrix collapses 8192x8192 -> 8192x512 (repeat-16 structure).
// GEMM via V_WMMA_F32_16X16X4_F32 (exact fp32); codebook staged to LDS with
// the Tensor Data Mover (tensor_load_to_lds + s_wait_tensorcnt).
// ---------------------------------------------------------------------------

#define N_PTS   8192     // 32*32*8 flattened latent vectors
#define K_CODE  512      // codebook entries
#define DIMV    64
#define EPSV    0.01f
#define INV_EPS 100.0f

#define CB_PHASE_ROWS 128          // codebook rows staged per LDS phase (32 KB)
#define ROWS_PER_BLOCK 128         // 8 waves x 16 rows

typedef __attribute__((ext_vector_type(2))) float v2f;
typedef __attribute__((ext_vector_type(8))) float v8f;
typedef __attribute__((ext_vector_type(4))) unsigned int v4u;
typedef __attribute__((ext_vector_type(4))) int v4i;
typedef __attribute__((ext_vector_type(8))) int v8i;

// ---- workspace layout (float offsets) -------------------------------------
#define WS_ZFLAT   0                         // [8192][64]
#define WS_COST    (WS_ZFLAT + N_PTS*DIMV)   // [8192][512]
#define WS_ZNORM   (WS_COST + N_PTS*K_CODE)  // [8192]
#define WS_PHILAT  (WS_ZNORM + N_PTS)        // [8192]
#define WS_CNORM   (WS_PHILAT + N_PTS)       // [512]
#define WS_PHIC    (WS_CNORM + K_CODE)       // [512]
#define WS_LSE1    (WS_PHIC + K_CODE)        // [8192]
#define WS_LSE2    (WS_LSE1 + N_PTS)         // [512]
#define WS_COUNTS  (WS_LSE2 + K_CODE)        // [512] (as uint)

// ---------------------------------------------------------------------------
__global__ __launch_bounds__(512) void zero_counts_kernel(unsigned* counts) {
    counts[threadIdx.x] = 0u;
}

// transpose z [8,64,32,32] -> z_flat [n][d] with n=(h*32+w)*8+b, plus ||z||^2
// and phi_lat[n] = z_n . kan2_w + kan2_b
__global__ __launch_bounds__(64) void prep_z_kernel(
    const float* __restrict__ z, const float* __restrict__ kan2_w,
    const float* __restrict__ kan2_b, float* __restrict__ zf,
    float* __restrict__ znorm, float* __restrict__ phi_lat) {
    const int n = blockIdx.x;
    const int d = threadIdx.x;
    const int b = n & 7, hw = n >> 3;
    const float val = z[b * 65536 + d * 1024 + hw];
    zf[n * DIMV + d] = val;
    __shared__ float s1[64], s2[64];
    s1[d] = val * val;
    s2[d] = val * kan2_w[d];
    __syncthreads();
    for (int off = 32; off > 0; off >>= 1) {
        if (d < off) { s1[d] += s1[d + off]; s2[d] += s2[d + off]; }
        __syncthreads();
    }
    if (d == 0) { znorm[n] = s1[0]; phi_lat[n] = s2[0] + kan2_b[0]; }
}

// ||c_j||^2 and phi_code[j] = c_j . kan1_w + kan1_b
__global__ __launch_bounds__(64) void prep_cb_kernel(
    const float* __restrict__ cb, const float* __restrict__ kan1_w,
    const float* __restrict__ kan1_b, float* __restrict__ cnorm,
    float* __restrict__ phiC) {
    const int j = blockIdx.x;
    const int d = threadIdx.x;
    const float val = cb[j * DIMV + d];
    __shared__ float s1[64], s2[64];
    s1[d] = val * val;
    s2[d] = val * kan1_w[d];
    __syncthreads();
    for (int off = 32; off > 0; off >>= 1) {
        if (d < off) { s1[d] += s1[d + off]; s2[d] += s2[d + off]; }
        __syncthreads();
    }
    if (d == 0) { cnorm[j] = s1[0]; phiC[j] = s2[0] + kan1_b[0]; }
}

// --- Tensor Data Mover: stage `CB_PHASE_ROWS` codebook rows into LDS at offset 0 -----
// D# per CDNA5 ISA. Only instruction emission matters
// here (compile-only loop); descriptor fields follow the documented bit
// layout: group0 = {count=1 | lds_addr=0 | global_addr | type=2},
// group1 = {mask=0, data_size=4B, tensor_dim0=64, tensor_dim1=rows,
//           tile_dim0=64, tile_dim1=rows, tensor_dim0_stride=64}.
__device__ __forceinline__ void tdm_load_codebook_phase(const float* gsrc) {
#if __has_builtin(__builtin_amdgcn_tensor_load_to_lds)
    const unsigned long long ga = (unsigned long long)(const void*)gsrc;
    v4u g0;
    g0.x = 1u;                                        // count=1, user mode
    g0.y = 0u;                                        // lds_addr = 0 (sole LDS alloc)
    g0.z = (unsigned int)(ga & 0xffffffffull);        // global_addr[31:0]
    g0.w = (unsigned int)((ga >> 32) & 0x01ffffffull) // global_addr[56:32]
         | 0x80000000u;                               // type = 2 ("image")
    v8i g1;
    g1[0] = 0x00020000;                               // data_size=2 (4 bytes)
    g1[1] = (int)(DIMV << 16);                        // tensor_dim0[15:0] = 64
    g1[2] = (int)(CB_PHASE_ROWS << 16);               // tensor_dim1[15:0] = 128
    g1[3] = (int)(DIMV << 16);                        // tile_dim0 = 64
    g1[4] = (int)CB_PHASE_ROWS;                       // tile_dim1 = 128
    g1[5] = (int)DIMV;                                // tensor_dim0_stride = 64
    g1[6] = 0;
    g1[7] = 0;
    v4i gz = {0, 0, 0, 0};
#if defined(__clang_major__) && (__clang_major__ >= 23)
    v8i gz8 = {0, 0, 0, 0, 0, 0, 0, 0};
    __builtin_amdgcn_tensor_load_to_lds(g0, g1, gz, gz, gz8, 0);
#else
    __builtin_amdgcn_tensor_load_to_lds(g0, g1, gz, gz, 0);
#endif
    __builtin_amdgcn_s_wait_tensorcnt(0);
#else
    (void)gsrc;   // fallback handled by caller
#endif
}

// costC[n][j] = ||z_n||^2 + ||c_j||^2 - 2 z_n.c_j.
// Block = 8 waves = 128 z rows x all 512 codebook columns. A fragments are
// preloaded once per wave (16 x float2 = 32 VGPRs) and reused across all 32
// column tiles; B fragments come from the TDM-staged LDS codebook.
__global__ __launch_bounds__(256) void cost_wmma_kernel(
    const float* __restrict__ zf, const float* __restrict__ cb,
    const float* __restrict__ znorm, const float* __restrict__ cnorm,
    float* __restrict__ costC) {
    __shared__ float lds_cb[CB_PHASE_ROWS * DIMV];    // 32 KB, LDS offset 0

    const int lane = threadIdx.x & 31;
    const int wave = threadIdx.x >> 5;                // 0..7
    const int half = lane >> 4;                       // 0/1
    const int l16  = lane & 15;
    const int row0 = blockIdx.x * ROWS_PER_BLOCK + wave * 16;

    // A fragment set: lane holds z_flat[row0+l16][4*kk + 2*half + {0,1}]
    const float* arow = zf + (row0 + l16) * DIMV + 2 * half;
    v2f afrag[DIMV / 4];
#pragma unroll
    for (int kk = 0; kk < DIMV / 4; ++kk)
        afrag[kk] = *(const v2f*)(arow + 4 * kk);

    for (int p = 0; p < K_CODE / CB_PHASE_ROWS; ++p) {   // 4 LDS phases
        const float* gsrc = cb + p * CB_PHASE_ROWS * DIMV;
#if __has_builtin(__builtin_amdgcn_tensor_load_to_lds)
        if (wave == 0) tdm_load_codebook_phase(gsrc);    // TDM, EXEC-independent
#else
        for (int i = threadIdx.x; i < CB_PHASE_ROWS * DIMV; i += 256)
            lds_cb[i] = gsrc[i];
#endif
        __syncthreads();

        for (int jt = 0; jt < CB_PHASE_ROWS / 16; ++jt) {
            const int col0 = p * CB_PHASE_ROWS + jt * 16;
            // B fragment: lane/VGPR r holds cb[col0+l16][4*kk + 2*half + r]
            const float* brow = &lds_cb[(jt * 16 + l16) * DIMV + 2 * half];
            v8f acc = {0.f, 0.f, 0.f, 0.f, 0.f, 0.f, 0.f, 0.f};
#pragma unroll
            for (int kk = 0; kk < DIMV / 4; ++kk) {
                v2f b = *(const v2f*)(brow + 4 * kk);  // ds_load_b64
                acc = __builtin_amdgcn_wmma_f32_16x16x4_f32(
                    /*neg_a=*/false, afrag[kk], /*neg_b=*/false, b,
                    /*c_mod=*/(short)0, acc,
                    /*reuse_a=*/false, /*reuse_b=*/false);
            }
            // D layout: VGPR v, lanes 0-15 -> M=v, lanes 16-31 -> M=v+8
            const float cn = cnorm[col0 + l16];
#pragma unroll
            for (int v = 0; v < 8; ++v) {
                const int m = v + 8 * half;
                const float cost = znorm[row0 + m] + cn - 2.0f * acc[v];
                costC[(row0 + m) * K_CODE + col0 + l16] = cost;
            }
        }
        __syncthreads();   // phase done before LDS is overwritten
    }
}

// Per-row (per z_n): argmin (first-occurrence tie-break), LSE for loss1
// (repeat-16 folded: -log 512), z_q scatter, bincount.
__global__ __launch_bounds__(32) void row_reduce_kernel(
    const float* __restrict__ costC, const float* __restrict__ phiC,
    const float* __restrict__ zf, const float* __restrict__ cb,
    float* __restrict__ lse1, unsigned* __restrict__ counts,
    float* __restrict__ zq_out) {
    const int n = blockIdx.x;
    const int lane = threadIdx.x;

    float e[16];
    float emax = -3.402823e38f;
    float cmin = 3.402823e38f;
    int jmin = 0x7fffffff;
#pragma unroll
    for (int t = 0; t < 16; ++t) {
        const int j = lane + 32 * t;
        const float c = costC[n * K_CODE + j];
        const float ev = (phiC[j] - c) * INV_EPS;
        e[t] = ev;
        emax = fmaxf(emax, ev);
        if (c < cmin) { cmin = c; jmin = j; }   // increasing j: keeps first
    }
    for (int m = 16; m >= 1; m >>= 1)
        emax = fmaxf(emax, __shfl_xor(emax, m, 32));
    float s = 0.0f;
#pragma unroll
    for (int t = 0; t < 16; ++t) s += expf(e[t] - emax);
    for (int m = 16; m >= 1; m >>= 1) s += __shfl_xor(s, m, 32);
    for (int m = 16; m >= 1; m >>= 1) {
        const float oc = __shfl_xor(cmin, m, 32);
        const int   oj = __shfl_xor(jmin, m, 32);
        if (oc < cmin || (oc == cmin && oj < jmin)) { cmin = oc; jmin = oj; }
    }
    if (lane == 0) {
        lse1[n] = emax + logf(s) - logf(512.0f);
        atomicAdd(&counts[jmin], 1u);
    }
    const int b = n & 7, hw = n >> 3;
#pragma unroll
    for (int t = 0; t < 2; ++t) {
        const int d = lane + 32 * t;
        const float zv = zf[n * DIMV + d];
        const float cv = cb[jmin * DIMV + d];
        zq_out[b * 65536 + d * 1024 + hw] = zv + (cv - zv);
    }
}

// Per-column (per codeword j): lse2[j] = logsumexp_n((phi_lat[n]-cost)/eps)
__global__ __launch_bounds__(256) void col_reduce_kernel(
    const float* __restrict__ costC, const float* __restrict__ phi_lat,
    float* __restrict__ lse2) {
    const int j = blockIdx.x;
    const int t = threadIdx.x;
    float e[32];
    float emax = -3.402823e38f;
#pragma unroll
    for (int i = 0; i < 32; ++i) {
        const int n = t + 256 * i;
        const float ev = (phi_lat[n] - costC[n * K_CODE + j]) * INV_EPS;
        e[i] = ev;
        emax = fmaxf(emax, ev);
    }
    __shared__ float sred[256];
    sred[t] = emax;
    __syncthreads();
    for (int off = 128; off > 0; off >>= 1) {
        if (t < off) sred[t] = fmaxf(sred[t], sred[t + off]);
        __syncthreads();
    }
    const float M = sred[0];
    __syncthreads();
    float s = 0.0f;
#pragma unroll
    for (int i = 0; i < 32; ++i) s += expf(e[i] - M);
    sred[t] = s;
    __syncthreads();
    for (int off = 128; off > 0; off >>= 1) {
        if (t < off) sred[t] += sred[t + off];
        __syncthreads();
    }
    if (t == 0) lse2[j] = M + logf(sred[0]);
}

// Deterministic single-block finalize: sums partial arrays, entropy, loss.
__global__ __launch_bounds__(512) void finalize_kernel(
    const float* __restrict__ lse1, const float* __restrict__ lse2,
    const float* __restrict__ phiC, const float* __restrict__ phi_lat,
    const unsigned* __restrict__ counts, float* __restrict__ out) {
    const int t = threadIdx.x;
    __shared__ float sA[512], sB[512], sC[512], sD[512], sE[512];
    float a = 0.0f, d = 0.0f;
    for (int n = t; n < N_PTS; n += 512) { a += lse1[n]; d += phi_lat[n]; }
    const float em = (float)counts[t] / (float)N_PTS;
    sA[t] = a;
    sB[t] = lse2[t];
    sC[t] = phiC[t];
    sD[t] = d;
    sE[t] = em * logf(em + 1e-10f);
    __syncthreads();
    for (int off = 256; off > 0; off >>= 1) {
        if (t < off) {
            sA[t] += sA[t + off]; sB[t] += sB[t + off]; sC[t] += sC[t + off];
            sD[t] += sD[t + off]; sE[t] += sE[t + off];
        }
        __syncthreads();
    }
    if (t == 0) {
        const float loss1 = (-EPSV / (float)N_PTS) * sA[0] + sC[0] / 512.0f;
        const float loss2 = -EPSV * logf(1.0f / (float)N_PTS)
                          + (-EPSV / 512.0f) * sB[0]
                          + sD[0] / (float)N_PTS;
        out[N_PTS * DIMV + 0] = 0.25f * (loss1 + loss2);   // loss
        out[N_PTS * DIMV + 1] = expf(-sE[0]);              // perplexity
    }
}

// ---------------------------------------------------------------------------
extern "C" void kernel_launch(void* const* d_in, const int* in_sizes, int n_in,
                              void* d_out, int out_size, void* d_ws, size_t ws_size,
                              hipStream_t stream) {
    const float* z       = (const float*)d_in[0];  // [8,64,32,32]
    const float* cb      = (const float*)d_in[1];  // [512,64]
    // d_in[2] = codebook_weight: unused in the fixed_weight path
    const float* kan1_w  = (const float*)d_in[3];  // [64]
    const float* kan1_b  = (const float*)d_in[4];  // [1]
    const float* kan2_w  = (const float*)d_in[5];  // [64]
    const float* kan2_b  = (const float*)d_in[6];  // [1]
    float* out = (float*)d_out;                    // [524288 zq | loss | perp]

    float* ws = (float*)d_ws;
    float* zf      = ws + WS_ZFLAT;
    float* costC   = ws + WS_COST;
    float* znorm   = ws + WS_ZNORM;
    float* phi_lat = ws + WS_PHILAT;
    float* cnorm   = ws + WS_CNORM;
    float* phiC    = ws + WS_PHIC;
    float* lse1    = ws + WS_LSE1;
    float* lse2    = ws + WS_LSE2;
    unsigned* counts = (unsigned*)(ws + WS_COUNTS);

    zero_counts_kernel<<<1, 512, 0, stream>>>(counts);
    prep_z_kernel<<<N_PTS, 64, 0, stream>>>(z, kan2_w, kan2_b, zf, znorm, phi_lat);
    prep_cb_kernel<<<K_CODE, 64, 0, stream>>>(cb, kan1_w, kan1_b, cnorm, phiC);
    cost_wmma_kernel<<<N_PTS / ROWS_PER_BLOCK, 256, 0, stream>>>(
        zf, cb, znorm, cnorm, costC);
    row_reduce_kernel<<<N_PTS, 32, 0, stream>>>(costC, phiC, zf, cb, lse1,
                                                counts, out);
    col_reduce_kernel<<<K_CODE, 256, 0, stream>>>(costC, phi_lat, lse2);
    finalize_kernel<<<1, 512, 0, stream>>>(lse1, lse2, phiC, phi_lat, counts, out);
}